// SumTaskGNN_60662118089064
// MI455X (gfx1250) — compile-verified
//
#include <hip/hip_runtime.h>

typedef __attribute__((ext_vector_type(2))) float v2f;
typedef __attribute__((ext_vector_type(8))) float v8f;

#define D_IN 128
#define D_HID 128
#define D_OUT 10
#define NUM_GRAPHS 64

// ---------------------------------------------------------------------------
// Kernel 1: scatter-add edges: agg[dst[e]] += x[src[e]]
// One wave32 per edge; each lane handles 4 consecutive floats (float4 = 16B,
// so a wave reads a full 512B row coalesced). Atomic adds are fire-and-forget
// f32 adds (no return) -> GLOBAL_ATOMIC_ADD_F32, tracked with STOREcnt.
// ---------------------------------------------------------------------------
__global__ void edge_scatter_kernel(const float* __restrict__ x,
                                    const int* __restrict__ src,
                                    const int* __restrict__ dst,
                                    float* __restrict__ agg,
                                    int nEdges) {
    long long tid  = (long long)blockIdx.x * blockDim.x + threadIdx.x;
    int edge = (int)(tid >> 5);
    int lane = (int)(tid & 31);
    if (edge >= nEdges) return;
    int s = src[edge];
    int d = dst[edge];
    const float4 v = *(const float4*)(x + (size_t)s * D_IN + lane * 4);
    float* base = agg + (size_t)d * D_IN + lane * 4;
    unsafeAtomicAdd(base + 0, v.x);
    unsafeAtomicAdd(base + 1, v.y);
    unsafeAtomicAdd(base + 2, v.z);
    unsafeAtomicAdd(base + 3, v.w);
}

// ---------------------------------------------------------------------------
// Kernel 2: h = relu(agg @ W_rel^T + x @ W_root^T), fused with
// sums[batch[i]] += h[i]  (global_mean_pool numerator) and counts.
//
// Uses V_WMMA_F32_16X16X4_F32 (full f32 precision, matches reference dtype).
// Block = 256 threads = 8 waves. Block owns a 16-node M-tile; wave w owns
// output dims [16w, 16w+16). K-loop: 128/4 = 32 WMMA steps per weight matrix
// (the two GEMMs are a K=256 concat into the same accumulator).
//
// VGPR layouts (ISA 7.12.2):
//   A (16x4 f32):  lane L -> row M=L%16, VGPR j holds K = kb + j + 2*(L/16)
//   B (4x16 f32):  lane L -> col N=L%16, VGPR j holds K = kb + j + 2*(L/16)
//                  with B[k][n] = W[n][k] (row-major W -> contiguous b64 load)
//   C/D (16x16):   c[i] at lane L -> (M = i + 8*(L/16), N = L%16)
// EXEC is all-1s at every WMMA (grid is exact: 50000 % 16 == 0, but guards
// below keep loads in-range anyway without masking the WMMA itself).
// ---------------------------------------------------------------------------
__global__ void gnn_wmma_kernel(const float* __restrict__ agg,
                                const float* __restrict__ x,
                                const float* __restrict__ Wrel,
                                const float* __restrict__ Wroot,
                                const int*   __restrict__ batch,
                                float* __restrict__ sums,
                                float* __restrict__ counts,
                                int nNodes) {
    const int nodeBase = blockIdx.x * 16;
    const int wave     = threadIdx.x >> 5;   // 0..7 -> N tile
    const int lane     = threadIdx.x & 31;
    const int half     = lane >> 4;          // 0 or 1 (K sub-pair select)
    const int lmod     = lane & 15;
    const int nBase    = wave * 16;

    int myNode = nodeBase + lmod;
    int aRowIdx = myNode < nNodes ? myNode : (nNodes - 1);   // clamp (no-op here)

    const float* arow  = agg   + (size_t)aRowIdx * D_IN + 2 * half;
    const float* xrow  = x     + (size_t)aRowIdx * D_IN + 2 * half;
    const float* brel  = Wrel  + (size_t)(nBase + lmod) * D_IN + 2 * half;
    const float* broot = Wroot + (size_t)(nBase + lmod) * D_IN + 2 * half;

    v8f c = {};

    #pragma unroll 4
    for (int kb = 0; kb < D_IN; kb += 4) {
        v2f a = *(const v2f*)(arow + kb);
        v2f b = *(const v2f*)(brel + kb);
        c = __builtin_amdgcn_wmma_f32_16x16x4_f32(
                /*neg_a=*/false, a, /*neg_b=*/false, b,
                /*c_mod=*/(short)0, c, /*reuse_a=*/false, /*reuse_b=*/false);
    }
    #pragma unroll 4
    for (int kb = 0; kb < D_IN; kb += 4) {
        v2f a = *(const v2f*)(xrow + kb);
        v2f b = *(const v2f*)(broot + kb);
        c = __builtin_amdgcn_wmma_f32_16x16x4_f32(
                false, a, false, b, (short)0, c, false, false);
    }

    // Fused relu + mean-pool numerator: c[i] = h[nodeBase + i + 8*half][nBase + lmod]
    #pragma unroll
    for (int i = 0; i < 8; ++i) {
        int node = nodeBase + i + 8 * half;
        if (node < nNodes) {
            float v = c[i] > 0.0f ? c[i] : 0.0f;
            int g = batch[node];
            unsafeAtomicAdd(&sums[(size_t)g * D_HID + nBase + lmod], v);
        }
    }
    // Node counts per graph: first wave, one lane per node of the tile.
    if (wave == 0 && lane < 16) {
        int node = nodeBase + lane;
        if (node < nNodes) {
            unsafeAtomicAdd(&counts[batch[node]], 1.0f);
        }
    }
}

// ---------------------------------------------------------------------------
// Kernel 3: out[g][o] = (1/max(counts[g],1)) * dot(sums[g], W_out[o])
// Tiny: 64*10 = 640 outputs, one block.
// ---------------------------------------------------------------------------
__global__ void finalize_kernel(const float* __restrict__ sums,
                                const float* __restrict__ counts,
                                const float* __restrict__ Wout,
                                float* __restrict__ out) {
    int t = threadIdx.x;
    if (t >= NUM_GRAPHS * D_OUT) return;
    int g = t / D_OUT;
    int o = t % D_OUT;
    float cnt = counts[g];
    cnt = cnt > 1.0f ? cnt : 1.0f;
    const float* srow = sums + (size_t)g * D_HID;
    const float* wrow = Wout + (size_t)o * D_HID;
    float acc = 0.0f;
    #pragma unroll 8
    for (int k = 0; k < D_HID; ++k) acc += srow[k] * wrow[k];
    out[t] = acc / cnt;
}

extern "C" void kernel_launch(void* const* d_in, const int* in_sizes, int n_in,
                              void* d_out, int out_size, void* d_ws, size_t ws_size,
                              hipStream_t stream) {
    const float* x          = (const float*)d_in[0];
    const int*   edge_index = (const int*)  d_in[1];
    const int*   batch      = (const int*)  d_in[2];
    const float* Wrel       = (const float*)d_in[3];
    const float* Wroot      = (const float*)d_in[4];
    const float* Wout       = (const float*)d_in[5];
    float*       out        = (float*)d_out;

    const int nEdges = in_sizes[1] / 2;   // edge_index is [2, E]
    const int nNodes = in_sizes[2];

    // Workspace layout: agg [nNodes*128] | sums [64*128] | counts [64]
    float* agg    = (float*)d_ws;
    float* sums   = agg + (size_t)nNodes * D_IN;
    float* counts = sums + (size_t)NUM_GRAPHS * D_HID;
    size_t zeroBytes = ((size_t)nNodes * D_IN + NUM_GRAPHS * D_HID + NUM_GRAPHS)
                       * sizeof(float);
    hipMemsetAsync(d_ws, 0, zeroBytes, stream);

    // 1) Edge scatter-add: one wave per edge.
    {
        long long threads = (long long)nEdges * 32;
        int blocks = (int)((threads + 255) / 256);
        edge_scatter_kernel<<<blocks, 256, 0, stream>>>(
            x, edge_index, edge_index + nEdges, agg, nEdges);
    }

    // 2) WMMA node transform + fused pooling accumulation.
    {
        int mTiles = (nNodes + 15) / 16;   // 3125 for 50000 nodes
        gnn_wmma_kernel<<<mTiles, 256, 0, stream>>>(
            agg, x, Wrel, Wroot, batch, sums, counts, nNodes);
    }

    // 3) Tiny final GEMM + mean divide.
    finalize_kernel<<<1, NUM_GRAPHS * D_OUT, 0, stream>>>(sums, counts, Wout, out);
}